// PoolAttentionForJE_45792941310016
// MI455X (gfx1250) — compile-verified
//
#include <hip/hip_runtime.h>
#include <hip/hip_bf16.h>
#include <math.h>
#include <stdint.h>

// Problem constants (match reference).
#define BATCH 32
#define SEQ   4096
#define VDIM  128
#define NSEG  (SEQ/2 + 2)     // 2050
#define SSPLIT 8              // Gram accumulation split along S for occupancy

typedef __attribute__((ext_vector_type(2))) float v2f;
typedef __attribute__((ext_vector_type(8))) float v8f;
typedef __attribute__((ext_vector_type(4))) unsigned int u32x4;
typedef __attribute__((ext_vector_type(8))) int i32x8;
typedef __attribute__((ext_vector_type(4))) int i32x4;

// ---------------------------------------------------------------------------
// WMMA f32 16x16x4 fragment helpers (wave32).
// A (16x4, MxK): lanes 0-15 hold M=lane, K=kk+0 (x) / kk+1 (y);
//                lanes 16-31 hold M=lane-16, K=kk+2 (x) / kk+3 (y).
// B (4x16, KxN): lanes 0-15 hold N=lane,   K=kk+0 (x) / kk+1 (y);
//                lanes 16-31 hold N=lane-16, K=kk+2 (x) / kk+3 (y).
// C/D (16x16):   element r of v8f: M = r + 8*(lane>=16), N = lane%16.
// ---------------------------------------------------------------------------
__device__ __forceinline__ v2f frag_colmajor(const float* __restrict__ buf, int stride,
                                             int kk, int base, int l16, int half) {
  v2f r;
  r.x = buf[(kk + 2 * half)     * stride + base + l16];
  r.y = buf[(kk + 2 * half + 1) * stride + base + l16];
  return r;
}
__device__ __forceinline__ v2f frag_rowmajor(const float* __restrict__ buf, int stride,
                                             int kk, int base, int l16, int half) {
  v2f r;
  r.x = buf[(base + l16) * stride + kk + 2 * half];
  r.y = buf[(base + l16) * stride + kk + 2 * half + 1];
  return r;
}
__device__ __forceinline__ v8f wmma4(v2f a, v2f b, v8f c) {
  return __builtin_amdgcn_wmma_f32_16x16x4_f32(false, a, false, b, (short)0, c, false, false);
}

// ---------------------------------------------------------------------------
// Tensor Data Mover: 1-D contiguous tile load (global -> LDS), f32 elements.
// D# per CDNA5 ISA ch.8: group0 {count=1, lds_addr, global_addr, type=2},
// group1 {data_size=4B, tensor_dim0=tile_dim0=nelem, dim1=1}; groups 2/3 and
// the trailing group are unused (zero) for a 1-D tile.  This toolchain's
// builtin takes the 6-arg form (g0, g1, g2, g3, g4, cpol).  Issued by one
// wave; completion tracked with TENSORcnt.
// ---------------------------------------------------------------------------
__device__ __forceinline__ void tdm_load_1d(uint32_t lds_off, const void* gptr, uint32_t nelem) {
  uint64_t ga = (uint64_t)gptr;
  u32x4 g0;
  g0[0] = 1u;                                            // count=1, user descriptor
  g0[1] = lds_off;                                       // lds_addr (bytes)
  g0[2] = (uint32_t)ga;                                  // global_addr[31:0]
  g0[3] = (uint32_t)((ga >> 32) & 0x1FFFFFFu) | (2u << 30); // global_addr[56:32] | type=2
  i32x8 g1;
  g1[0] = (int)(2u << 16);                               // data_size=2 -> 4 bytes
  g1[1] = (int)((nelem & 0xFFFFu) << 16);                // tensor_dim0[15:0]
  g1[2] = (int)(((nelem >> 16) & 0xFFFFu) | (1u << 16)); // tensor_dim0[31:16], tensor_dim1=1
  g1[3] = (int)((nelem & 0xFFFFu) << 16);                // tile_dim0 = nelem (<= 65535)
  g1[4] = 0;                                             // tile_dim1/2 unused
  g1[5] = (int)nelem;                                    // tensor_dim0_stride[31:0]
  g1[6] = 0;
  g1[7] = 0;
  i32x4 z4 = {0, 0, 0, 0};
  i32x8 z8 = {0, 0, 0, 0, 0, 0, 0, 0};
  __builtin_amdgcn_tensor_load_to_lds(g0, g1, z4, z4, z8, 0);
}

// ---------------------------------------------------------------------------
// Kernel 0: sinusoidal positional encoding pe[S][V]
// ---------------------------------------------------------------------------
__global__ void pe_kernel(float* __restrict__ pe) {
  int t = blockIdx.x;            // 0..S-1
  int i = threadIdx.x;           // 0..63 (pairs)
  float freq = __expf((2.0f * i) * (-logf(10000.0f) / (float)VDIM));
  float ang = (float)t * freq;
  pe[t * VDIM + 2 * i]     = sinf(ang);
  pe[t * VDIM + 2 * i + 1] = cosf(ang);
}

// ---------------------------------------------------------------------------
// Kernel 1: zero segment accumulators (must be re-zeroed every call)
// ---------------------------------------------------------------------------
__global__ void zero_kernel(float* __restrict__ p, int n) {
  for (int i = blockIdx.x * blockDim.x + threadIdx.x; i < n; i += gridDim.x * blockDim.x)
    p[i] = 0.0f;
}

// ---------------------------------------------------------------------------
// Kernel 2: per-row rising-edge scan -> seg_id (inclusive cumsum), num_time_seq
// ---------------------------------------------------------------------------
__global__ __launch_bounds__(1024) void scan_kernel(const float* __restrict__ start,
                                                    int* __restrict__ seg_id,
                                                    int* __restrict__ num_out) {
  __shared__ int sc[1024];
  int b = blockIdx.x, tid = threadIdx.x;
  int j0 = tid * 4;
  const float* row = start + (size_t)b * SEQ;
  float s0 = row[j0], s1 = row[j0 + 1], s2 = row[j0 + 2], s3 = row[j0 + 3];
  float pv = (j0 == 0) ? 0.0f : row[j0 - 1];   // t==0: prev treated as < 0.5
  int g0 = s0 >= 0.5f, g1 = s1 >= 0.5f, g2 = s2 >= 0.5f, g3 = s3 >= 0.5f;
  int gp = pv >= 0.5f;
  int m0 = g0 & !gp, m1 = g1 & !g0, m2 = g2 & !g1, m3 = g3 & !g2;
  int c0 = m0, c1 = c0 + m1, c2 = c1 + m2, c3 = c2 + m3;
  sc[tid] = c3;
  __syncthreads();
  for (int off = 1; off < 1024; off <<= 1) {
    int v = (tid >= off) ? sc[tid - off] : 0;
    __syncthreads();
    sc[tid] += v;
    __syncthreads();
  }
  int excl = sc[tid] - c3;
  int* out = seg_id + (size_t)b * SEQ + j0;
  out[0] = excl + c0; out[1] = excl + c1; out[2] = excl + c2; out[3] = excl + c3;
  if (tid == 0) num_out[b] = sc[1023] + 1;
}

// ---------------------------------------------------------------------------
// Kernel 3: segment sums & counts of xseq = bert + 2*pe (atomic segment_sum)
// ---------------------------------------------------------------------------
__global__ void segsum_kernel(const float* __restrict__ bert, const float* __restrict__ pe,
                              const int* __restrict__ seg_id,
                              float* __restrict__ segsum, float* __restrict__ segcnt) {
  int g = blockIdx.x * 2 + (threadIdx.x >> 7);   // global token index
  int v = threadIdx.x & 127;
  int b = g / SEQ, t = g % SEQ;
  float val = bert[(size_t)g * VDIM + v] + 2.0f * pe[t * VDIM + v];
  int seg = seg_id[(size_t)b * SEQ + t];
  atomicAdd(&segsum[((size_t)b * NSEG + seg) * VDIM + v], val);
  if (v == 0) atomicAdd(&segcnt[(size_t)b * NSEG + seg], 1.0f);
}

// ---------------------------------------------------------------------------
// Kernel 4: partial Gram G = X^T X with X = xseq[b] (S x 128), f32 WMMA.
// grid (SSPLIT, B); block 1024 = 32 waves; each wave owns 2 of the 64 output
// 16x16 tiles.  64-token chunks are DMA'd into LDS by the Tensor Data Mover
// (double-buffered, overlapping the WMMA phase); a fixup pass adds 2*pe.
// ---------------------------------------------------------------------------
__global__ __launch_bounds__(1024) void gram_kernel(const float* __restrict__ bert,
                                                    const float* __restrict__ pe,
                                                    float* __restrict__ gpart) {
  __shared__ float Xs[2][64 * VDIM];            // 2 x 32 KB double buffer
  int sp = blockIdx.x, b = blockIdx.y;
  int tid = threadIdx.x, lane = tid & 31, wave = tid >> 5;
  int l16 = lane & 15, half = lane >> 4;
  int ti0 = 2 * wave, ti1 = 2 * wave + 1;
  int m0 = (ti0 >> 3) * 16, n0 = (ti0 & 7) * 16;
  int m1 = (ti1 >> 3) * 16, n1 = (ti1 & 7) * 16;
  v8f acc0 = {}, acc1 = {};
  const float* Xb = bert + (size_t)b * SEQ * VDIM;
  int tbase = sp * (SEQ / SSPLIT);
  const int nchunks = (SEQ / SSPLIT) / 64;      // 8
  // LDS generic address: bits [31:0] are the LDS byte offset (shared aperture).
  uint32_t lds0 = (uint32_t)(uintptr_t)&Xs[0][0];
  uint32_t lds1 = (uint32_t)(uintptr_t)&Xs[1][0];
  if (wave == 0)
    tdm_load_1d(lds0, Xb + (size_t)tbase * VDIM, 64 * VDIM);
  for (int c = 0; c < nchunks; ++c) {
    int p = c & 1;
    if (wave == 0) __builtin_amdgcn_s_wait_tensorcnt(0);  // buffer p landed
    __syncthreads();                                       // publish to all waves
    // fixup: xseq = bert + 2*pe (in place in LDS)
    int tstart = tbase + c * 64;
    for (int e = tid; e < 64 * VDIM; e += 1024) {
      int tt = e >> 7, vv = e & 127;
      Xs[p][e] += 2.0f * pe[(tstart + tt) * VDIM + vv];
    }
    __syncthreads();
    // kick DMA for the next chunk; overlaps the WMMA phase below.
    // (buffer p^1's previous readers retired at the last iteration's barrier)
    if (wave == 0 && c + 1 < nchunks)
      tdm_load_1d(p ? lds0 : lds1, Xb + (size_t)(tbase + (c + 1) * 64) * VDIM, 64 * VDIM);
    const float* Xp = &Xs[p][0];
    for (int kk = 0; kk < 64; kk += 4) {
      v2f a0 = frag_colmajor(Xp, VDIM, kk, m0, l16, half);
      v2f b0 = frag_colmajor(Xp, VDIM, kk, n0, l16, half);
      acc0 = wmma4(a0, b0, acc0);
      v2f a1 = frag_colmajor(Xp, VDIM, kk, m1, l16, half);
      v2f b1 = frag_colmajor(Xp, VDIM, kk, n1, l16, half);
      acc1 = wmma4(a1, b1, acc1);
    }
    __syncthreads();                                       // retire reads of buffer p
  }
  float* gp = gpart + ((size_t)b * SSPLIT + sp) * (VDIM * VDIM);
  for (int r = 0; r < 8; ++r) {
    gp[(m0 + r + 8 * half) * VDIM + n0 + l16] = acc0[r];
    gp[(m1 + r + 8 * half) * VDIM + n1 + l16] = acc1[r];
  }
}

// Kernel 5: reduce partial Grams
__global__ void gred_kernel(const float* __restrict__ gpart, float* __restrict__ G) {
  int idx = blockIdx.x * blockDim.x + threadIdx.x;     // B * 128*128 threads
  int b = idx / (VDIM * VDIM), j = idx % (VDIM * VDIM);
  float s = 0.0f;
  for (int sp = 0; sp < SSPLIT; ++sp)
    s += gpart[((size_t)b * SSPLIT + sp) * (VDIM * VDIM) + j];
  G[idx] = s;
}

// ---------------------------------------------------------------------------
// Kernel 6: per-batch kv = (Wk G Wv^T + (Wk m)bv^T + bk(Wv m)^T + S bk bv^T)/sqrt(S)
// ---------------------------------------------------------------------------
#define LPAD 132
__global__ __launch_bounds__(256) void kv_kernel(const float* __restrict__ G,
                                                 const float* __restrict__ segsum,
                                                 const float* __restrict__ Wk,
                                                 const float* __restrict__ bk,
                                                 const float* __restrict__ Wv,
                                                 const float* __restrict__ bv,
                                                 float* __restrict__ kv) {
  __shared__ float Gs[VDIM * VDIM];       // column reads -> no pad needed
  __shared__ float T1s[VDIM * LPAD];      // row-major A reads -> padded
  __shared__ float mld[VDIM], km[VDIM], wm[VDIM];
  int b = blockIdx.x, tid = threadIdx.x, lane = tid & 31, wave = tid >> 5;
  int l16 = lane & 15, half = lane >> 4;
  const float* Gb = G + (size_t)b * VDIM * VDIM;
  for (int e = tid; e < VDIM * VDIM; e += 256) Gs[e] = Gb[e];
  if (tid < VDIM) {
    const float* ss = segsum + (size_t)b * NSEG * VDIM + tid;
    float s = 0.0f;
    for (int seg = 0; seg < NSEG; ++seg) s += ss[(size_t)seg * VDIM];
    mld[tid] = s;
  }
  __syncthreads();
  if (tid < VDIM) {
    float a = 0.0f, c = 0.0f;
    for (int k = 0; k < VDIM; ++k) { a += Wk[tid * VDIM + k] * mld[k];
                                     c += Wv[tid * VDIM + k] * mld[k]; }
    km[tid] = a; wm[tid] = c;
  }
  // Phase 1: T1 = Wk @ G   (A = Wk row-major from global, B = Gs)
  int mbase = wave * 16;
  for (int ni = 0; ni < 8; ++ni) {
    int nbase = ni * 16;
    v8f acc = {};
    for (int kk = 0; kk < VDIM; kk += 4) {
      v2f a; a.x = Wk[(mbase + l16) * VDIM + kk + 2 * half];
             a.y = Wk[(mbase + l16) * VDIM + kk + 2 * half + 1];
      v2f bf = frag_colmajor(Gs, VDIM, kk, nbase, l16, half);
      acc = wmma4(a, bf, acc);
    }
    for (int r = 0; r < 8; ++r)
      T1s[(mbase + r + 8 * half) * LPAD + nbase + l16] = acc[r];
  }
  __syncthreads();
  // Phase 2: kv = T1 @ Wv^T + rank-1 corrections, pre-scaled by 1/sqrt(S)
  float* kvb = kv + (size_t)b * VDIM * VDIM;
  const float inv_sqrt_s = 1.0f / 64.0f;   // 1/sqrt(4096)
  for (int ni = 0; ni < 8; ++ni) {
    int nbase = ni * 16;
    v8f acc = {};
    for (int kk = 0; kk < VDIM; kk += 4) {
      v2f a = frag_rowmajor(T1s, LPAD, kk, mbase, l16, half);
      v2f bf; bf.x = Wv[(nbase + l16) * VDIM + kk + 2 * half];
              bf.y = Wv[(nbase + l16) * VDIM + kk + 2 * half + 1];
      acc = wmma4(a, bf, acc);
    }
    int w = nbase + l16;
    for (int r = 0; r < 8; ++r) {
      int v = mbase + r + 8 * half;
      float val = acc[r] + km[v] * bv[w] + bk[v] * wm[w] + (float)SEQ * bk[v] * bv[w];
      kvb[v * VDIM + w] = val * inv_sqrt_s;
    }
  }
}

// ---------------------------------------------------------------------------
// Kernel 7 (generic M x 128 @ 128 x 128 WMMA GEMM), used twice:
//   q_seg    = (segsum / max(cnt,1)) @ Wq^T + bq        (transB=1, cnt!=null)
//   attn_seg = q_seg @ kv_scaled                        (transB=0, per-batch W)
// ---------------------------------------------------------------------------
__global__ __launch_bounds__(256) void gemm_kernel(const float* __restrict__ A, size_t aBS,
                                                   const float* __restrict__ W, size_t wBS,
                                                   const float* __restrict__ bias,
                                                   const float* __restrict__ cnt,
                                                   float* __restrict__ Out, size_t oBS,
                                                   int M, int transB) {
  __shared__ float As[16 * LPAD];
  __shared__ float Bs[VDIM * VDIM];       // column reads -> no pad needed
  int b = blockIdx.y, tid = threadIdx.x, lane = tid & 31, wave = tid >> 5;
  int l16 = lane & 15, half = lane >> 4;
  int row0 = blockIdx.x * 16;
  const float* Ab = A + (size_t)b * aBS;
  const float* Wb = W + (size_t)b * wBS;
  for (int e = tid; e < 16 * 32; e += 256) {          // float4-staged A strip
    int rr = e >> 5, q = e & 31;
    int row = row0 + rr;
    float4 v4 = make_float4(0.f, 0.f, 0.f, 0.f);
    if (row < M) {
      v4 = reinterpret_cast<const float4*>(Ab + (size_t)row * VDIM)[q];
      if (cnt) {
        float inv = 1.0f / fmaxf(cnt[(size_t)b * NSEG + row], 1.0f);
        v4.x *= inv; v4.y *= inv; v4.z *= inv; v4.w *= inv;
      }
    }
    float* dst = &As[rr * LPAD + q * 4];
    dst[0] = v4.x; dst[1] = v4.y; dst[2] = v4.z; dst[3] = v4.w;
  }
  if (transB) {
    for (int e = tid; e < VDIM * VDIM; e += 256) {
      int k = e >> 7, n = e & 127;
      Bs[e] = Wb[n * VDIM + k];
    }
  } else {
    for (int e = tid; e < VDIM * 32; e += 256) {      // float4-staged direct copy
      int k = e >> 5, q = e & 31;
      reinterpret_cast<float4*>(&Bs[k * VDIM])[q] =
          reinterpret_cast<const float4*>(Wb + k * VDIM)[q];
    }
  }
  __syncthreads();
  int nbase = wave * 16;
  v8f acc = {};
  for (int kk = 0; kk < VDIM; kk += 4) {
    v2f a  = frag_rowmajor(As, LPAD, kk, 0, l16, half);
    v2f bf = frag_colmajor(Bs, VDIM, kk, nbase, l16, half);
    acc = wmma4(a, bf, acc);
  }
  int col = nbase + l16;
  float bb = bias ? bias[col] : 0.0f;
  float* Ob = Out + (size_t)b * oBS;
  for (int r = 0; r < 8; ++r) {
    int row = row0 + r + 8 * half;
    if (row < M) Ob[(size_t)row * VDIM + col] = acc[r] + bb;
  }
}

// ---------------------------------------------------------------------------
// Kernel 8: emit — out[:, :, 0:128] = attn_seg gathered by seg_id,
//                  out[:, :, 128:256] = bert + pe.  Pure bandwidth, float4.
// ---------------------------------------------------------------------------
__global__ void emit_kernel(const float* __restrict__ bert, const float* __restrict__ pe,
                            const int* __restrict__ seg_id,
                            const float* __restrict__ attnseg, float* __restrict__ out) {
  int idx = blockIdx.x * blockDim.x + threadIdx.x;   // B*S*32 threads (float4 over V)
  int q = idx & 31;
  int t = (idx >> 5) & (SEQ - 1);
  int b = idx >> 17;
  int g = b * SEQ + t;
  int seg = seg_id[g];
  const float4 a4 = reinterpret_cast<const float4*>(attnseg + ((size_t)b * NSEG + seg) * VDIM)[q];
  const float4 x4 = reinterpret_cast<const float4*>(bert + (size_t)g * VDIM)[q];
  const float4 p4 = reinterpret_cast<const float4*>(pe + (size_t)t * VDIM)[q];
  float4* orow = reinterpret_cast<float4*>(out + (size_t)g * (2 * VDIM));
  orow[q] = a4;
  orow[32 + q] = make_float4(x4.x + p4.x, x4.y + p4.y, x4.z + p4.z, x4.w + p4.w);
}

// ---------------------------------------------------------------------------
// Host-side orchestration
// ---------------------------------------------------------------------------
extern "C" void kernel_launch(void* const* d_in, const int* in_sizes, int n_in,
                              void* d_out, int out_size, void* d_ws, size_t ws_size,
                              hipStream_t stream) {
  const float* bert = (const float*)d_in[0];
  const float* start = (const float*)d_in[1];
  const float* Wq = (const float*)d_in[2];
  const float* bq = (const float*)d_in[3];
  const float* Wk = (const float*)d_in[4];
  const float* bk = (const float*)d_in[5];
  const float* Wv = (const float*)d_in[6];
  const float* bv = (const float*)d_in[7];
  float* out = (float*)d_out;

  // Workspace layout (float elements)
  float* ws = (float*)d_ws;
  const size_t N_PE     = (size_t)SEQ * VDIM;                 // 524288
  const size_t N_SEGID  = (size_t)BATCH * SEQ;                // 131072 (ints)
  const size_t N_SEGSUM = (size_t)BATCH * NSEG * VDIM;        // 8396800
  const size_t N_SEGCNT = (size_t)BATCH * NSEG;               // 65600
  const size_t N_MAT    = (size_t)BATCH * VDIM * VDIM;        // 524288
  float* pe      = ws;                                        size_t off = N_PE;
  int*   seg_id  = (int*)(ws + off);                          off += N_SEGID;
  float* segsum  = ws + off;                                  off += N_SEGSUM;
  float* segcnt  = ws + off;                                  off += N_SEGCNT;   // contiguous w/ segsum
  float* qseg    = ws + off;                                  off += N_SEGSUM;
  float* attnseg = ws + off;                                  off += N_SEGSUM;
  float* gpart   = ws + off;                                  off += N_MAT * SSPLIT;
  float* G       = ws + off;                                  off += N_MAT;
  float* kv      = ws + off;                                  off += N_MAT;

  // num_time_seq tail of d_out (int32 bits after the B*S*2V float tensor)
  int* num_out = (int*)d_out + (size_t)BATCH * SEQ * 2 * VDIM;

  pe_kernel<<<SEQ, 64, 0, stream>>>(pe);
  zero_kernel<<<4096, 256, 0, stream>>>(segsum, (int)(N_SEGSUM + N_SEGCNT));
  scan_kernel<<<BATCH, 1024, 0, stream>>>(start, seg_id, num_out);
  segsum_kernel<<<BATCH * SEQ / 2, 256, 0, stream>>>(bert, pe, seg_id, segsum, segcnt);
  gram_kernel<<<dim3(SSPLIT, BATCH), 1024, 0, stream>>>(bert, pe, gpart);
  gred_kernel<<<(BATCH * VDIM * VDIM) / 256, 256, 0, stream>>>(gpart, G);
  kv_kernel<<<BATCH, 256, 0, stream>>>(G, segsum, Wk, bk, Wv, bv, kv);
  const int tilesM = (NSEG + 15) / 16;   // 129
  gemm_kernel<<<dim3(tilesM, BATCH), 256, 0, stream>>>(
      segsum, (size_t)NSEG * VDIM, Wq, 0, bq, segcnt,
      qseg, (size_t)NSEG * VDIM, NSEG, /*transB=*/1);
  gemm_kernel<<<dim3(tilesM, BATCH), 256, 0, stream>>>(
      qseg, (size_t)NSEG * VDIM, kv, (size_t)VDIM * VDIM, nullptr, nullptr,
      attnseg, (size_t)NSEG * VDIM, NSEG, /*transB=*/0);
  emit_kernel<<<(BATCH * SEQ * 32) / 256, 256, 0, stream>>>(bert, pe, seg_id, attnseg, out);
}